// FreeEnergy_38646115729716
// MI455X (gfx1250) — compile-verified
//
#include <hip/hip_runtime.h>
#include <math.h>

// CDNA5 / gfx1250, wave32. All matrix math in f32 via V_WMMA_F32_16X16X4_F32
// (precision required by Cholesky/logdet path; only f32 WMMA shape).

typedef __attribute__((ext_vector_type(2))) float v2f;
typedef __attribute__((ext_vector_type(8))) float v8f;

#define NMAX 512
#define EPSR 1e-6f
#define WPB 8     // waves per block
#define TPB 256   // threads per block (8 wave32)

// LDS ops from one wave are processed in order (DScnt in-order); this wait
// makes cross-lane LDS results visible and fences compiler reordering.
__device__ __forceinline__ void wave_sync() {
  asm volatile("s_wait_dscnt 0x0" ::: "memory");
}

__device__ __forceinline__ float wave_sum(float v) {
#pragma unroll
  for (int o = 16; o >= 1; o >>= 1) v += __shfl_xor(v, o, 32);
  return v;
}

// C/D 16x16 f32 fragment: lane half h, col l: vgpr r holds D[r + 8*h][l].
__device__ __forceinline__ v8f ident_frag(int lane) {
  int hh = lane >> 4, l = lane & 15;
  v8f f;
#pragma unroll
  for (int r = 0; r < 8; ++r) f[r] = ((r + 8 * hh) == l) ? 1.0f : 0.0f;
  return f;
}

// acc += A(16x16) * B(16x16), both row-major stride 16 (LDS or global).
// 4 chained v_wmma_f32_16x16x4_f32.
// A-frag: lanes 0-15 row l, v0=K(k),v1=K(k+1); lanes 16-31 v0=K(k+2),v1=K(k+3).
// B-frag mirrored: v0: lanes0-15=B[k][l], lanes16-31=B[k+2][l]; v1: k+1/k+3.
__device__ __forceinline__ v8f mm16(const float* A, float as, const float* B,
                                    v8f acc, int lane) {
  int hh = lane >> 4, l = lane & 15;
#pragma unroll
  for (int k = 0; k < 16; k += 4) {
    int ka = k + 2 * hh;
    v2f a, b;
    a.x = A[l * 16 + ka] * as;
    a.y = A[l * 16 + ka + 1] * as;
    b.x = B[ka * 16 + l];
    b.y = B[(ka + 1) * 16 + l];
    acc = __builtin_amdgcn_wmma_f32_16x16x4_f32(false, a, false, b, (short)0,
                                                acc, false, false);
  }
  return acc;
}

// acc += A * Bt^T  (B operand is Bt transposed; fragment element
// B[ka][l] = Bt[l][ka], i.e. A-style addressing on Bt).
__device__ __forceinline__ v8f mm16_bt(const float* A, const float* Bt,
                                       v8f acc, int lane) {
  int hh = lane >> 4, l = lane & 15;
#pragma unroll
  for (int k = 0; k < 16; k += 4) {
    int ka = k + 2 * hh;
    v2f a, b;
    a.x = A[l * 16 + ka];
    a.y = A[l * 16 + ka + 1];
    b.x = Bt[l * 16 + ka];
    b.y = Bt[l * 16 + ka + 1];
    acc = __builtin_amdgcn_wmma_f32_16x16x4_f32(false, a, false, b, (short)0,
                                                acc, false, false);
  }
  return acc;
}

__device__ __forceinline__ void store_frag(float* D, v8f acc, int lane) {
  int hh = lane >> 4, l = lane & 15;
#pragma unroll
  for (int r = 0; r < 8; ++r) D[(r + 8 * hh) * 16 + l] = acc[r];
}

// R = exp(16*Y): degree-8 Taylor of exp(Y) (Horner, identity-frag accumulator)
// followed by 4 squarings. Caller passes Y = X/16; ||X||<~3 so ||Y||<~0.2,
// Taylor error ~0.2^9/9! ~ 1e-12.
__device__ __forceinline__ void expm16(const float* Y, float* R, int lane) {
#pragma unroll
  for (int e = 0; e < 8; ++e) {
    int idx = lane * 8 + e, r = idx >> 4, c = idx & 15;
    R[idx] = ((r == c) ? 1.0f : 0.0f) + Y[idx] * (1.0f / 8.0f);
  }
  wave_sync();
#pragma unroll 1
  for (int m = 7; m >= 1; --m) {
    v8f acc = ident_frag(lane);
    acc = mm16(Y, 1.0f / (float)m, R, acc, lane);
    store_frag(R, acc, lane);
    wave_sync();
  }
#pragma unroll 1
  for (int s = 0; s < 4; ++s) {
    v8f acc = {};
    acc = mm16(R, 1.0f, R, acc, lane);
    store_frag(R, acc, lane);
    wave_sync();
  }
}

// Wave-cooperative in-place Cholesky of SPD 16x16 in LDS (lower+diag valid).
__device__ __forceinline__ void chol16(float* W, int lane) {
#pragma unroll 1
  for (int c = 0; c < 16; ++c) {
    float s = W[c * 16 + c];
    for (int k = 0; k < c; ++k) {
      float v = W[c * 16 + k];
      s -= v * v;
    }
    float d = sqrtf(fmaxf(s, 1e-30f));
    if (lane == c) W[c * 16 + c] = d;
    if (lane > c && lane < 16) {
      float s2 = W[lane * 16 + c];
      for (int k = 0; k < c; ++k) s2 -= W[lane * 16 + k] * W[c * 16 + k];
      W[lane * 16 + c] = s2 / d;
    }
    wave_sync();
  }
}

// ||column col of Lp^{-1} Lq||^2 (Lq entries above diag treated as 0).
__device__ __forceinline__ float trace_col(const float* Lp, const float* Lq,
                                           int col) {
  float y[16];
  float ssq = 0.0f;
#pragma unroll
  for (int r = 0; r < 16; ++r) {
    float s = (r >= col) ? Lq[r * 16 + col] : 0.0f;
#pragma unroll
    for (int k = 0; k < r; ++k) s -= Lp[r * 16 + k] * y[k];
    float yr = s / Lp[r * 16 + r];
    y[r] = yr;
    ssq += yr * yr;
  }
  return ssq;
}

__device__ __forceinline__ float quad_solve(const float* Lp, const float* d) {
  float y[16];
  float q = 0.0f;
#pragma unroll
  for (int r = 0; r < 16; ++r) {
    float s = d[r];
#pragma unroll
    for (int k = 0; k < r; ++k) s -= Lp[r * 16 + k] * y[k];
    float yr = s / Lp[r * 16 + r];
    y[r] = yr;
    q += yr * yr;
  }
  return q;
}

// ---------------- Phase 1: per-i exp(+/-X), chol(Sq), self KL ----------------
__global__ void __launch_bounds__(TPB) fe_phase1(
    const float* __restrict__ muq, const float* __restrict__ Sq,
    const float* __restrict__ mup, const float* __restrict__ Sp,
    const float* __restrict__ phi, const float* __restrict__ gen,
    float* __restrict__ Pg, float* __restrict__ Mg, float* __restrict__ Lqg,
    float* __restrict__ ldqg, float* __restrict__ klself, int N) {
  __shared__ float Ys[WPB][256];
  __shared__ float Rs[WPB][256];
  __shared__ float dls[WPB][16];
  const int w = threadIdx.x >> 5, lane = threadIdx.x & 31;
  const int i = blockIdx.x * WPB + w;
  if (i >= N) return;  // wave-uniform; no block syncs in this kernel
  float* Y = Ys[w];
  float* R = Rs[w];
  float* dl = dls[w];

  // Y = (phi_i . generators) / 16
  const float p0 = phi[i * 3 + 0], p1 = phi[i * 3 + 1], p2 = phi[i * 3 + 2];
#pragma unroll
  for (int e = 0; e < 8; ++e) {
    int idx = lane * 8 + e;
    float x = p0 * gen[idx] + p1 * gen[256 + idx] + p2 * gen[512 + idx];
    Y[idx] = x * (1.0f / 16.0f);
  }
  wave_sync();
  expm16(Y, R, lane);  // R = exp(+X)
#pragma unroll
  for (int e = 0; e < 8; ++e) Pg[i * 256 + lane * 8 + e] = R[lane * 8 + e];
#pragma unroll
  for (int e = 0; e < 8; ++e) Y[lane * 8 + e] = -Y[lane * 8 + e];
  wave_sync();
  expm16(Y, R, lane);  // R = exp(-X)
#pragma unroll
  for (int e = 0; e < 8; ++e) Mg[i * 256 + lane * 8 + e] = R[lane * 8 + e];
  wave_sync();

  // L_q = chol(Sigma_q + eps I) in Y; logdet_q
#pragma unroll
  for (int e = 0; e < 8; ++e) {
    int idx = lane * 8 + e, r = idx >> 4, c = idx & 15;
    Y[idx] = Sq[i * 256 + idx] + ((r == c) ? EPSR : 0.0f);
  }
  wave_sync();
  chol16(Y, lane);
  float ldq = 2.0f * wave_sum((lane < 16) ? logf(Y[lane * 17]) : 0.0f);
#pragma unroll
  for (int e = 0; e < 8; ++e) {
    int idx = lane * 8 + e, r = idx >> 4, c = idx & 15;
    Lqg[i * 256 + idx] = (c <= r) ? Y[idx] : 0.0f;  // zero strict upper
  }
  if (lane == 0) ldqg[i] = ldq;

  // L_p = chol(Sigma_p + eps I) in R; KL(q_i || p_i)
#pragma unroll
  for (int e = 0; e < 8; ++e) {
    int idx = lane * 8 + e, r = idx >> 4, c = idx & 15;
    R[idx] = Sp[i * 256 + idx] + ((r == c) ? EPSR : 0.0f);
  }
  if (lane < 16) dl[lane] = mup[i * 16 + lane] - muq[i * 16 + lane];
  wave_sync();
  chol16(R, lane);
  float ldp = 2.0f * wave_sum((lane < 16) ? logf(R[lane * 17]) : 0.0f);
  float part = 0.0f;
  if (lane < 16) part = trace_col(R, Y, lane);
  else if (lane == 16) part = quad_solve(R, dl);
  float tq = wave_sum(part);
  if (lane == 0) klself[i] = fmaxf(0.5f * (tq - 16.0f + ldp - ldq), 0.0f);
}

// --------- Phase 2: one block per row i; pairwise transported KL ------------
__global__ void __launch_bounds__(TPB) fe_phase2(
    const float* __restrict__ muq, const float* __restrict__ Sq,
    const float* __restrict__ Pg, const float* __restrict__ Mg,
    const float* __restrict__ Lqg, const float* __restrict__ ldqg,
    float* __restrict__ erow, int N) {
  __shared__ float Pi[256], Lqi[256], mui[16];
  __shared__ float klrow[NMAX];
  __shared__ float red[TPB];
  __shared__ float Oms[WPB][256], Wts[WPB][256], dls[WPB][16];
  const int i = blockIdx.x;
  const int t = threadIdx.x;
  const int w = t >> 5, lane = t & 31;
  Pi[t] = Pg[i * 256 + t];
  Lqi[t] = Lqg[i * 256 + t];
  if (t < 16) mui[t] = muq[i * 16 + t];
  __syncthreads();
  const float ldq = ldqg[i];
  float* Om = Oms[w];
  float* W = Wts[w];
  float* dl = dls[w];

  for (int j = w; j < N; j += WPB) {  // trip count uniform per wave
    if (j == i) {                     // wave-uniform branch
      if (lane == 0) klrow[j] = 0.0f;
      continue;
    }
    // Omega = P_i * M_j  (B fragments streamed from L2-resident global)
    v8f acc = {};
    acc = mm16(Pi, 1.0f, Mg + j * 256, acc, lane);
    store_frag(Om, acc, lane);
    wave_sync();
    // delta = Omega * mu_j - mu_i
    if (lane < 16) {
      float s = 0.0f;
#pragma unroll
      for (int c = 0; c < 16; ++c) s += Om[lane * 16 + c] * muq[j * 16 + c];
      dl[lane] = s - mui[lane];
    }
    // OS = Omega * Sigma_j
    v8f acc2 = {};
    acc2 = mm16(Om, 1.0f, Sq + j * 256, acc2, lane);
    store_frag(W, acc2, lane);
    wave_sync();
    // Sigma_t = OS * Omega^T
    v8f acc3 = {};
    acc3 = mm16_bt(W, Om, acc3, lane);
    store_frag(W, acc3, lane);
    wave_sync();
    // symmetrize + eps I (reads fully precede writes; DS in-order per wave)
    float tv[8];
#pragma unroll
    for (int e = 0; e < 8; ++e) {
      int idx = lane * 8 + e, r = idx >> 4, c = idx & 15;
      tv[e] = 0.5f * (W[idx] + W[c * 16 + r]) + ((r == c) ? EPSR : 0.0f);
    }
    wave_sync();
#pragma unroll
    for (int e = 0; e < 8; ++e) W[lane * 8 + e] = tv[e];
    wave_sync();
    chol16(W, lane);
    float ldp = 2.0f * wave_sum((lane < 16) ? logf(W[lane * 17]) : 0.0f);
    float part = 0.0f;
    if (lane < 16) part = trace_col(W, Lqi, lane);
    else if (lane == 16) part = quad_solve(W, dl);
    float tq = wave_sum(part);
    float kl = fmaxf(0.5f * (tq - 16.0f + ldp - ldq), 0.0f);
    if (lane == 0) klrow[j] = kl;
  }
  __syncthreads();

  // Row softmax(-kl) weighted sum (kappa=1), diagonal excluded.
  float mn = 3.0e38f;
  for (int j = t; j < N; j += TPB)
    if (j != i) mn = fminf(mn, klrow[j]);
  red[t] = mn;
  __syncthreads();
  for (int s = TPB / 2; s > 0; s >>= 1) {
    if (t < s) red[t] = fminf(red[t], red[t + s]);
    __syncthreads();
  }
  mn = red[0];
  __syncthreads();
  float z = 0.0f, wsum = 0.0f;
  for (int j = t; j < N; j += TPB)
    if (j != i) {
      float e = __expf(mn - klrow[j]);
      z += e;
      wsum += e * klrow[j];
    }
  red[t] = z;
  __syncthreads();
  for (int s = TPB / 2; s > 0; s >>= 1) {
    if (t < s) red[t] += red[t + s];
    __syncthreads();
  }
  z = red[0];
  __syncthreads();
  red[t] = wsum;
  __syncthreads();
  for (int s = TPB / 2; s > 0; s >>= 1) {
    if (t < s) red[t] += red[t + s];
    __syncthreads();
  }
  if (t == 0) erow[i] = red[0] / z;
}

// --------- Phase 3: deterministic fixed-tree final reduction ----------------
__global__ void __launch_bounds__(256) fe_finalize(const float* __restrict__ klself,
                                                   const float* __restrict__ erow,
                                                   float* __restrict__ out, int N) {
  __shared__ float r1[256], r2[256];
  const int t = threadIdx.x;
  float s1 = 0.0f, s2 = 0.0f;
  for (int j = t; j < N; j += 256) {
    s1 += klself[j];
    s2 += erow[j];
  }
  r1[t] = s1;
  r2[t] = s2;
  __syncthreads();
  for (int s = 128; s > 0; s >>= 1) {
    if (t < s) {
      r1[t] += r1[t + s];
      r2[t] += r2[t + s];
    }
    __syncthreads();
  }
  if (t == 0) {
    float eself = r1[0], ebelief = r2[0];
    out[0] = eself + ebelief;  // lambda_self = lambda_belief = 1
    out[1] = eself;
    out[2] = ebelief;
    out[3] = 0.0f;             // lambda_prior = 0
  }
}

extern "C" void kernel_launch(void* const* d_in, const int* in_sizes, int n_in,
                              void* d_out, int out_size, void* d_ws,
                              size_t ws_size, hipStream_t stream) {
  const float* muq = (const float*)d_in[0];
  const float* Sq = (const float*)d_in[1];
  const float* mup = (const float*)d_in[2];
  const float* Sp = (const float*)d_in[3];
  const float* phi = (const float*)d_in[4];
  const float* gen = (const float*)d_in[5];
  int N = in_sizes[0] / 16;
  if (N > NMAX) N = NMAX;

  float* wsf = (float*)d_ws;
  size_t o = 0;
  float* Pg = wsf + o;     o += (size_t)N * 256;  // exp(+X_i)
  float* Mg = wsf + o;     o += (size_t)N * 256;  // exp(-X_i)
  float* Lqg = wsf + o;    o += (size_t)N * 256;  // chol(Sigma_q + eps)
  float* ldqg = wsf + o;   o += (size_t)N;        // logdet_q
  float* klself = wsf + o; o += (size_t)N;        // per-i self KL
  float* erow = wsf + o;   o += (size_t)N;        // per-row belief energy

  const int blocks1 = (N + WPB - 1) / WPB;
  fe_phase1<<<blocks1, TPB, 0, stream>>>(muq, Sq, mup, Sp, phi, gen, Pg, Mg,
                                         Lqg, ldqg, klself, N);
  fe_phase2<<<N, TPB, 0, stream>>>(muq, Sq, Pg, Mg, Lqg, ldqg, erow, N);
  fe_finalize<<<1, 256, 0, stream>>>(klself, erow, (float*)d_out, N);
}